// MeanModel_11166914970000
// MI455X (gfx1250) — compile-verified
//
#include <hip/hip_runtime.h>

// Problem constants (from reference): x[B, L, K, C], reduce over L, broadcast back.
#define B_      8
#define L_      4096
#define KC_     2048            // K*C = 32*64, contiguous innermost
#define NCHUNK_ 16              // L split into 16 chunks of 256 for parallelism
#define LCH_    (L_ / NCHUNK_)  // 256
#define TOT_    (B_ * KC_)      // 16384 reduction columns total

typedef __attribute__((ext_vector_type(2))) float v2f;
typedef __attribute__((ext_vector_type(4))) float v4f;
typedef __attribute__((ext_vector_type(8))) float v8f;
typedef __attribute__((ext_vector_type(4))) int   v4i;

// ---------------------------------------------------------------------------
// Phase 1: streaming partial reduction over an L-chunk.
// Each thread owns 4 contiguous channels -> 128-bit coalesced loads; a wave
// covers 512 contiguous bytes per L row. Loads are non-temporal (single-touch
// 512 MB stream vs 192 MB L2). Unroll-8 keeps ~8 KB in flight per wave
// (~16 MB machine-wide across 2048 waves) to cover HBM latency at 23.3 TB/s.
// ---------------------------------------------------------------------------
__global__ __launch_bounds__(128) void mm_reduce_partial(
    const float* __restrict__ x, const int* __restrict__ mask,
    float* __restrict__ pS, float* __restrict__ pC)
{
    const int col = (blockIdx.x * 128 + threadIdx.x) * 4;   // [0, 2048)
    const int b   = blockIdx.y;
    const int ch  = blockIdx.z;

    const size_t base = ((size_t)b * L_ + (size_t)ch * LCH_) * KC_ + (size_t)col;
    const v4f* __restrict__ xp = (const v4f*)x;
    const v4i* __restrict__ mp = (const v4i*)mask;
    size_t i4 = base >> 2;
    const size_t str = KC_ >> 2;   // one L row = 512 x 16B

    float s0 = 0.f, s1 = 0.f, s2 = 0.f, s3 = 0.f;
    int   c0 = 0,   c1 = 0,   c2 = 0,   c3 = 0;

    for (int l = 0; l < LCH_; l += 8) {
        // Speculative prefetch 32 rows ahead (gfx1250 global_prefetch_b8);
        // past-end translation failures are silently dropped per ISA.
        __builtin_prefetch(xp + (i4 + 32 * str), 0, 3);
        __builtin_prefetch(mp + (i4 + 32 * str), 0, 3);
#pragma unroll
        for (int u = 0; u < 8; ++u) {
            const v4f xv = __builtin_nontemporal_load(xp + i4);
            const v4i mv = __builtin_nontemporal_load(mp + i4);
            i4 += str;
            s0 += mv.x ? xv.x : 0.0f;  c0 += mv.x;
            s1 += mv.y ? xv.y : 0.0f;  c1 += mv.y;
            s2 += mv.z ? xv.z : 0.0f;  c2 += mv.z;
            s3 += mv.w ? xv.w : 0.0f;  c3 += mv.w;
        }
    }

    const size_t o = (size_t)ch * TOT_ + (size_t)b * KC_ + (size_t)col;
    v4f sv; sv.x = s0; sv.y = s1; sv.z = s2; sv.w = s3;
    v4f cv; cv.x = (float)c0; cv.y = (float)c1; cv.z = (float)c2; cv.w = (float)c3;
    *(v4f*)(pS + o) = sv;   // partials are re-read in phase 2: keep RT policy
    *(v4f*)(pC + o) = cv;
}

// ---------------------------------------------------------------------------
// Phase 2: cross-chunk reduction via V_WMMA_F32_16X16X4_F32 with all-ones A:
//   D[m][n] = sum_k A[m][k]*B[k][n] + C[m][n] = sum_k B[k][n] + C[m][n]
// Each wave reduces 16 chunks x 16 columns with 4 WMMAs per matrix (sum+cnt).
// Counts are small integers held in f32 -> exact. Ones-A makes the result
// independent of the exact K<->lane packing; N<->lane follows the C/D layout.
// ---------------------------------------------------------------------------
__global__ __launch_bounds__(256) void mm_finalize(
    const float* __restrict__ pS, const float* __restrict__ pC,
    float* __restrict__ mean)
{
    const int gtid = blockIdx.x * 256 + threadIdx.x;
    const int wave = gtid >> 5;                    // 1024 waves total
    const int lane = threadIdx.x & 31;             // wave32
    const int hi   = lane >> 4;
    const int cg   = wave * 16 + (lane & 15);      // global column [0, 16384)

    v2f a; a[0] = 1.0f; a[1] = 1.0f;               // ones A (layout-proof)
    v8f accS = {};
    v8f accC = {};

#if __has_builtin(__builtin_amdgcn_wmma_f32_16x16x4_f32)
    for (int k0 = 0; k0 < NCHUNK_; k0 += 4) {
        const int r0 = k0 + 2 * hi;                // B layout: lanes 0-15 rows {r0,r0+1}, lanes 16-31 rows {r0+2,r0+3}
        v2f bS, bC;
        bS[0] = pS[(size_t)(r0    ) * TOT_ + cg];
        bS[1] = pS[(size_t)(r0 + 1) * TOT_ + cg];
        bC[0] = pC[(size_t)(r0    ) * TOT_ + cg];
        bC[1] = pC[(size_t)(r0 + 1) * TOT_ + cg];
        accS = __builtin_amdgcn_wmma_f32_16x16x4_f32(
                   false, a, false, bS, (short)0, accS, false, false);
        accC = __builtin_amdgcn_wmma_f32_16x16x4_f32(
                   false, a, false, bC, (short)0, accC, false, false);
    }
    const float s = accS[0];                       // row M=0 on lanes 0-15
    const float c = accC[0];
#else
    // Fallback (keeps compile green if the builtin is absent on this toolchain)
    float s = 0.f, c = 0.f;
    for (int k = 0; k < NCHUNK_; ++k) {
        s += pS[(size_t)k * TOT_ + cg];
        c += pC[(size_t)k * TOT_ + cg];
    }
#endif

    if (lane < 16) {
        mean[cg] = (c > 0.0f) ? (s / c) : 0.0f;    // cnt>0 => divide by cnt (>=1)
    }
}

// ---------------------------------------------------------------------------
// Phase 3: broadcast mean back to [B, L, K, C]. The 64 KB mean table is read
// with default (RT) policy so it stays L2-resident across 16M re-reads; the
// 256 MB output stream is written non-temporally (single-touch, > L2).
// ---------------------------------------------------------------------------
__global__ __launch_bounds__(256) void mm_broadcast(
    const float* __restrict__ mean, float* __restrict__ out)
{
    const size_t t    = (size_t)blockIdx.x * 256 + threadIdx.x;
    const size_t flat = t << 2;                     // float index
    const int col = (int)(flat & (size_t)(KC_ - 1));
    const int b   = (int)(flat >> 23);              // L*KC = 2^23
    const v4f mv = *(const v4f*)(mean + (size_t)b * KC_ + col);
    __builtin_nontemporal_store(mv, (v4f*)out + t);
}

// ---------------------------------------------------------------------------
extern "C" void kernel_launch(void* const* d_in, const int* in_sizes, int n_in,
                              void* d_out, int out_size, void* d_ws, size_t ws_size,
                              hipStream_t stream)
{
    (void)in_sizes; (void)n_in; (void)out_size; (void)ws_size;
    const float* x    = (const float*)d_in[0];
    const int*   mask = (const int*)d_in[1];
    float*       out  = (float*)d_out;

    // Workspace layout (needs 2*16*16384*4 + 16384*4 = ~2.11 MB):
    float* pS   = (float*)d_ws;                      // [NCHUNK][B*KC] partial sums
    float* pC   = pS + (size_t)NCHUNK_ * TOT_;       // [NCHUNK][B*KC] partial counts
    float* mean = pC + (size_t)NCHUNK_ * TOT_;       // [B*KC]

    // Phase 1: 512 blocks (4 col-groups x 8 batch x 16 L-chunks), 4 waves each.
    dim3 g1(KC_ / (128 * 4), B_, NCHUNK_);
    mm_reduce_partial<<<g1, 128, 0, stream>>>(x, mask, pS, pC);

    // Phase 2: 1024 waves / 8 waves per block = 128 blocks.
    mm_finalize<<<(TOT_ / 16) * 32 / 256, 256, 0, stream>>>(pS, pC, mean);

    // Phase 3: 67108864 floats / (256 threads * 4) = 65536 blocks.
    const size_t total = (size_t)B_ * L_ * KC_;
    mm_broadcast<<<(unsigned)(total / (256 * 4)), 256, 0, stream>>>(mean, out);
}